// YoloLoss_94489280638
// MI455X (gfx1250) — compile-verified
//
#include <hip/hip_runtime.h>
#include <math.h>

// ---------------------------------------------------------------------------
// YOLO-v1 loss on gfx1250 (MI455X). Memory-bound streaming reduction:
//   kernel 1: grid-stride over N*S*S cells, 5 f32 partial sums per thread,
//             wave reduction via V_WMMA_F32_16X16X4_F32 (ones-matrix trick),
//             cross-wave via LDS, per-block partials to workspace.
//   kernel 2: one block reduces block partials (again via WMMA wave-sum),
//             applies means/sqrt/lambda weights, writes the 6 outputs.
// ---------------------------------------------------------------------------

#define SGRID 7
#define NCLS  20
#define IMGF  448.0f
#define CELLF 64.0f          // 448 / 7
#define EPSF  1e-6f

typedef __attribute__((ext_vector_type(2))) float v2f;
typedef __attribute__((ext_vector_type(8))) float v8f;

// Full-wave (32 lane) f32 sum using one V_WMMA_F32_16X16X4_F32.
// A[m][k] holds each lane's value (a.y = 0), B = all-ones 4x16, C = 0:
//   D[m][n] = sum_k A[m][k]  ->  x_m + x_{m+16}   (rows m = 0..15)
// C/D layout: VGPR v of lanes 0-15 = row v, lanes 16-31 = row v+8, so
// sum(c[0..7]) gives rows 0..7 (low half) / 8..15 (high half); xor-16
// shuffle-add completes the 32-lane total in every lane. All f32, exact
// up to rounding. Requires EXEC all ones -> call with no divergence.
__device__ __forceinline__ float wave_sum32(float x) {
    v2f a; a.x = x;    a.y = 0.0f;
    v2f b; b.x = 1.0f; b.y = 1.0f;
    v8f c = {};
    c = __builtin_amdgcn_wmma_f32_16x16x4_f32(
            /*neg_a=*/false, a, /*neg_b=*/false, b,
            /*c_mod=*/(short)0, c, /*reuse_a=*/false, /*reuse_b=*/false);
    float s = ((c[0] + c[1]) + (c[2] + c[3])) + ((c[4] + c[5]) + (c[6] + c[7]));
    s += __shfl_xor(s, 16, 32);
    return s;
}

__device__ __forceinline__ float iou_vs(float bx, float by, float bw, float bh,
                                        float fi, float fj,
                                        float ta0, float ta1, float ta2, float ta3,
                                        float areaT) {
    float xc = (bx + fj) * CELLF;
    float yc = (by + fi) * CELLF;
    float hw = bw * (IMGF * 0.5f);
    float hh = bh * (IMGF * 0.5f);
    float a0 = xc - hw, a1 = yc - hh, a2 = xc + hw, a3 = yc + hh;
    float x1 = fmaxf(ta0, a0), y1 = fmaxf(ta1, a1);
    float x2 = fminf(ta2, a2), y2 = fminf(ta3, a3);
    float inter = fmaxf(x2 - x1, 0.0f) * fmaxf(y2 - y1, 0.0f);
    float areaP = (a2 - a0) * (a3 - a1);
    return inter / (areaT + areaP - inter + EPSF);
}

__global__ void __launch_bounds__(256)
yolo_partial(const float* __restrict__ pred,   // (N,7,7,30)
             const float* __restrict__ targ,   // (N,7,7,25)
             float* __restrict__ part,         // (gridDim.x, 5)
             int ncells) {
    float s_xy = 0.0f, s_wh = 0.0f, s_co = 0.0f, s_cn = 0.0f, s_cl = 0.0f;

    const int stride = gridDim.x * blockDim.x;
    for (int cell = blockIdx.x * blockDim.x + threadIdx.x; cell < ncells; cell += stride) {
        const int rem = cell % (SGRID * SGRID);
        const float fi = (float)(rem / SGRID);   // y-cell index
        const float fj = (float)(rem % SGRID);   // x-cell index

        const float* __restrict__ p = pred + (size_t)cell * 30;
        const float* __restrict__ t = targ + (size_t)cell * 25;

        // prediction cell base is 8-byte aligned (cell*120B) -> b64 loads
        float pv[30];
        const v2f* __restrict__ p2 = reinterpret_cast<const v2f*>(p);
        #pragma unroll
        for (int k = 0; k < 15; ++k) { v2f v = p2[k]; pv[2*k] = v.x; pv[2*k+1] = v.y; }
        float tv[25];
        #pragma unroll
        for (int k = 0; k < 25; ++k) tv[k] = t[k];

        // target abs box
        const float txc = (tv[0] + fj) * CELLF;
        const float tyc = (tv[1] + fi) * CELLF;
        const float thw = tv[2] * (IMGF * 0.5f);
        const float thh = tv[3] * (IMGF * 0.5f);
        const float ta0 = txc - thw, ta1 = tyc - thh, ta2 = txc + thw, ta3 = tyc + thh;
        const float areaT = (ta2 - ta0) * (ta3 - ta1);

        const float iou1 = iou_vs(pv[0], pv[1], pv[2], pv[3], fi, fj, ta0, ta1, ta2, ta3, areaT);
        const float iou2 = iou_vs(pv[5], pv[6], pv[7], pv[8], fi, fj, ta0, ta1, ta2, ta3, areaT);

        // reference: idx = 5 * (iou1 > iou2) -> picks SECOND box when iou1 wins
        const bool pick2 = (iou1 > iou2);
        const float b0 = pick2 ? pv[5] : pv[0];
        const float b1 = pick2 ? pv[6] : pv[1];
        const float b2 = pick2 ? pv[7] : pv[2];
        const float b3 = pick2 ? pv[8] : pv[3];
        const float b4 = pick2 ? pv[9] : pv[4];

        const float t4    = tv[4];
        const bool  noobj = (t4 == 0.0f);

        // xy loss (clip at 0, zero when no object)
        const float xh = noobj ? 0.0f : fmaxf(b0, 0.0f);
        const float yh = noobj ? 0.0f : fmaxf(b1, 0.0f);
        float dx = tv[0] - xh, dy = tv[1] - yh;
        s_xy = fmaf(dx, dx, s_xy); s_xy = fmaf(dy, dy, s_xy);

        // wh loss (sqrt applied to the global mean, not per-cell)
        const float wh0 = noobj ? 0.0f : fmaxf(b2, 0.0f);
        const float wh1 = noobj ? 0.0f : fmaxf(b3, 0.0f);
        float dw = tv[2] - wh0, dh = tv[3] - wh1;
        s_wh = fmaf(dw, dw, s_wh); s_wh = fmaf(dh, dh, s_wh);

        // confidence
        const float ch = fminf(fmaxf(b4, 0.0f), 1.0f);
        const float d_obj = t4 - (noobj ? 0.0f : ch);
        s_co = fmaf(d_obj, d_obj, s_co);
        const float d_no = t4 - (noobj ? ch : 0.0f);
        s_cn = fmaf(d_no, d_no, s_cn);

        // class loss
        #pragma unroll
        for (int k = 0; k < NCLS; ++k) {
            float d = tv[5 + k] - pv[10 + k];
            s_cl = fmaf(d, d, s_cl);
        }
    }

    // wave reduction via WMMA (all 256 threads reach here -> EXEC all ones)
    s_xy = wave_sum32(s_xy);
    s_wh = wave_sum32(s_wh);
    s_co = wave_sum32(s_co);
    s_cn = wave_sum32(s_cn);
    s_cl = wave_sum32(s_cl);

    __shared__ float red[8][5];
    const int wave = threadIdx.x >> 5;
    const int lane = threadIdx.x & 31;
    if (lane == 0) {
        red[wave][0] = s_xy; red[wave][1] = s_wh; red[wave][2] = s_co;
        red[wave][3] = s_cn; red[wave][4] = s_cl;
    }
    __syncthreads();
    if (threadIdx.x < 5) {
        float a = 0.0f;
        const int nw = blockDim.x >> 5;
        for (int w = 0; w < nw; ++w) a += red[w][threadIdx.x];
        part[(size_t)blockIdx.x * 5 + threadIdx.x] = a;
    }
}

__global__ void __launch_bounds__(256)
yolo_final(const float* __restrict__ part, int nblocks,
           float* __restrict__ out, int ncells) {
    float s[5] = {0.0f, 0.0f, 0.0f, 0.0f, 0.0f};
    for (int b = threadIdx.x; b < nblocks; b += blockDim.x) {
        #pragma unroll
        for (int q = 0; q < 5; ++q) s[q] += part[(size_t)b * 5 + q];
    }
    #pragma unroll
    for (int q = 0; q < 5; ++q) s[q] = wave_sum32(s[q]);

    __shared__ float red[8][5];
    const int wave = threadIdx.x >> 5;
    const int lane = threadIdx.x & 31;
    if (lane == 0) {
        #pragma unroll
        for (int q = 0; q < 5; ++q) red[wave][q] = s[q];
    }
    __syncthreads();
    if (threadIdx.x == 0) {
        float tot[5] = {0.0f, 0.0f, 0.0f, 0.0f, 0.0f};
        const int nw = blockDim.x >> 5;
        for (int w = 0; w < nw; ++w)
            for (int q = 0; q < 5; ++q) tot[q] += red[w][q];

        const float nc   = (float)ncells;         // N*49  (exact in f32)
        const float lxy  = tot[0] / (2.0f * nc);
        const float lwh  = sqrtf(tot[1] / (2.0f * nc) + EPSF);
        const float lco  = tot[2] / nc;
        const float lcn  = tot[3] / nc;
        const float lcl  = tot[4] / (20.0f * nc);
        const float loss = 5.0f * lxy + 5.0f * lwh + lco + 0.5f * lcn + lcl;

        out[0] = loss; out[1] = lxy; out[2] = lwh;
        out[3] = lco;  out[4] = lcn; out[5] = lcl;
    }
}

extern "C" void kernel_launch(void* const* d_in, const int* in_sizes, int n_in,
                              void* d_out, int out_size, void* d_ws, size_t ws_size,
                              hipStream_t stream) {
    const float* pred = (const float*)d_in[0];
    const float* targ = (const float*)d_in[1];
    float* out  = (float*)d_out;
    float* part = (float*)d_ws;

    const int ncells = in_sizes[0] / 30;          // N * S * S

    int nblocks = 512;                            // ~3 cells/thread at N=8192
    const size_t need = (size_t)nblocks * 5 * sizeof(float);
    if (need > ws_size) {
        nblocks = (int)(ws_size / (5 * sizeof(float)));
        if (nblocks < 1) nblocks = 1;
    }

    yolo_partial<<<nblocks, 256, 0, stream>>>(pred, targ, part, ncells);
    yolo_final<<<1, 256, 0, stream>>>(part, nblocks, out, ncells);
}